// GraphEmbedder_44598940402225
// MI455X (gfx1250) — compile-verified
//
#include <hip/hip_runtime.h>

typedef __attribute__((ext_vector_type(2))) float v2f;
typedef __attribute__((ext_vector_type(8))) float v8f;
typedef __attribute__((ext_vector_type(4))) unsigned int v4u;
typedef __attribute__((ext_vector_type(4))) int v4i;
typedef __attribute__((ext_vector_type(8))) int v8i;

#define EMB 128
#define NODES_PER_BLOCK 32
#define W_STRIDE 132    // padded LDS row stride (128 + 4 dwords TDM pad)
#define A_STRIDE 132

__device__ __forceinline__ float lrelu(float x) { return x > 0.0f ? x : 0.01f * x; }

// ---- Tensor Data Mover: 2D (or 1D when tile_d1==0) load, 4B elements,
// ---- LDS padding: +4 dwords after every 128 dwords (row stride 132).
__device__ __forceinline__ void tdm_load_padded(const void* gaddr, unsigned lds_off,
                                                unsigned tensor_d0, unsigned tensor_d1,
                                                unsigned tile_d0, unsigned tile_d1,
                                                unsigned stride0_elems) {
    unsigned long long ga = (unsigned long long)(uintptr_t)gaddr;
    v4u g0;
    g0[0] = 1u;                                   // count=1, no gather, user mode
    g0[1] = lds_off;                              // lds_addr (bytes)
    g0[2] = (unsigned)(ga & 0xFFFFFFFFu);         // global_addr[31:0]
    g0[3] = (unsigned)((ga >> 32) & 0x1FFFFFFu)   // global_addr[56:32]
          | (2u << 30);                           // type=2 ("image")
    v8i g1;
    g1[0] = (int)((2u << 16)      // data_size = 4 bytes
                | (1u << 20)      // pad_enable
                | (6u << 22)      // pad_interval: 128 dwords
                | (3u << 25));    // pad_amount: 4 dwords
    g1[1] = (int)(tensor_d0 << 16);                         // dim0[15:0] in [31:16]
    g1[2] = (int)((tensor_d0 >> 16) | (tensor_d1 << 16));   // dim0 hi | dim1 lo
    g1[3] = (int)((tensor_d1 >> 16) | (tile_d0 << 16));     // dim1 hi | tile_dim0
    g1[4] = (int)(tile_d1 & 0xFFFFu);                       // tile_dim1 (0 => 1D), tile_dim2=0
    g1[5] = (int)stride0_elems;                             // tensor_dim0_stride[31:0]
    g1[6] = 0;
    g1[7] = 0;
    v4i z4 = {0, 0, 0, 0};
    v8i z8 = {0, 0, 0, 0, 0, 0, 0, 0};
    __builtin_amdgcn_tensor_load_to_lds(g0, g1, z4, z4, z8, 0);
}

// -------- row_ptr[n] = lower_bound(edge_src, n), n in [0, nN] --------
__global__ void k_rowptr(const int* __restrict__ src, int nE, int nN, int* __restrict__ rp) {
    int n = blockIdx.x * blockDim.x + threadIdx.x;
    if (n > nN) return;
    int lo = 0, hi = nE;
    while (lo < hi) { int mid = (lo + hi) >> 1; if (src[mid] < n) lo = mid + 1; else hi = mid; }
    rp[n] = lo;
}

// -------- S[n] = sum of edge weights of node n --------
__global__ void k_nodewsum(const float* __restrict__ ew, const int* __restrict__ rp,
                           int nN, float* __restrict__ S) {
    int n = blockIdx.x * blockDim.x + threadIdx.x;
    if (n >= nN) return;
    float s = 0.0f;
    int e1 = rp[n + 1];
    for (int e = rp[n]; e < e1; ++e) s += ew[e];
    S[n] = s;
}

// -------- v[j] = sum_k w_nbweights[j][k] * lrelu(w_ew[k]); zero colsum/qconst --------
__global__ void k_precompute(const float* __restrict__ wnbw, const float* __restrict__ wew,
                             float* __restrict__ v, float* __restrict__ colsum,
                             float* __restrict__ qconst) {
    int j = threadIdx.x;  // 128 threads
    float acc = 0.0f;
    #pragma unroll 4
    for (int k = 0; k < EMB; ++k) acc += wnbw[j * EMB + k] * lrelu(wew[k]);
    v[j] = acc;
    colsum[j] = 0.0f;
    if (j == 0) *qconst = 0.0f;
}

// -------- iteration 1: emb = lrelu(base), base[n][j] = is_sel[n]*wsel[j] + S[n]*v[j] --------
__global__ void k_emb1(const int* __restrict__ feat, const float* __restrict__ S,
                       const float* __restrict__ wsel, const float* __restrict__ v,
                       float* __restrict__ emb, int nN) {
    int idx = blockIdx.x * blockDim.x + threadIdx.x;
    if (idx >= nN * EMB) return;
    int n = idx >> 7, j = idx & 127;
    emb[idx] = lrelu((float)feat[n] * wsel[j] + S[n] * v[j]);
}

// -------- main iteration: emb_out = lrelu(base + segsum(emb_in[dst]) @ W^T) --------
__global__ __launch_bounds__(128) void k_iter(
    const float* __restrict__ emb_in, float* __restrict__ emb_out,
    const int* __restrict__ rp, const int* __restrict__ dstv,
    const int* __restrict__ feat, const float* __restrict__ S,
    const float* __restrict__ wsel, const float* __restrict__ v,
    const float* __restrict__ W, int nN)
{
    extern __shared__ float smem[];
    float* Wl = smem;                        // [128][W_STRIDE] row-major padded (TDM)
    float* At = smem + EMB * W_STRIDE;       // [32][A_STRIDE]

    const int tid  = threadIdx.x;
    const int lane = tid & 31;
    const int wave = tid >> 5;
    const int node0 = blockIdx.x * NODES_PER_BLOCK;

    // Kick off the 64KB weight DMA; it overlaps the gather phase below.
    if (wave == 0)
        tdm_load_padded(W, 0u, EMB, EMB, EMB, EMB, EMB);

    // Phase A: neighbor aggregation; one wave per node, float4 per lane (128 floats/row)
    const float4* embv = (const float4*)emb_in;
    for (int t = 0; t < 8; ++t) {
        int m = wave * 8 + t;
        int n = node0 + m;
        float4 acc = make_float4(0.f, 0.f, 0.f, 0.f);
        if (n < nN) {
            int e1 = rp[n + 1];
            for (int e = rp[n]; e < e1; ++e) {
                int d = dstv[e];
                float4 x = embv[(size_t)d * (EMB / 4) + lane];
                acc.x += x.x; acc.y += x.y; acc.z += x.z; acc.w += x.w;
            }
        }
        *(float4*)(At + m * A_STRIDE + lane * 4) = acc;
    }
    if (wave == 0) __builtin_amdgcn_s_wait_tensorcnt(0);
    __syncthreads();

    // Phase B: 2 row-tiles x 8 col-tiles of 16x16, V_WMMA_F32_16X16X4_F32, K=128
    const int mrow  = lane & 15;   // M (A) / N (B) index within tile
    const int khalf = lane >> 4;   // K sub-offset selector
    const bool full = (node0 + NODES_PER_BLOCK) <= nN;
    for (int tt = wave; tt < 16; tt += 4) {
        int mt = tt & 1;
        int jt = tt >> 1;
        v8f c = {};
        const float* ap = At + (mt * 16 + mrow) * A_STRIDE + 2 * khalf;   // A[m][k..k+1]
        const float* bp = Wl + (jt * 16 + mrow) * W_STRIDE + 2 * khalf;   // W[n][k..k+1] = B^T
        #pragma unroll 8
        for (int k0 = 0; k0 < EMB; k0 += 4) {
            v2f a = *(const v2f*)(ap + k0);
            v2f b = *(const v2f*)(bp + k0);
            c = __builtin_amdgcn_wmma_f32_16x16x4_f32(false, a, false, b,
                                                      (short)0, c, false, false);
        }
        int j = jt * 16 + mrow;
        float wsj = wsel[j], vj = v[j];
        if (full) {
            #pragma unroll
            for (int r = 0; r < 8; ++r) {
                int n = node0 + mt * 16 + r + 8 * khalf;
                float x = c[r] + (float)feat[n] * wsj + S[n] * vj;
                emb_out[(size_t)n * EMB + j] = lrelu(x);
            }
        } else {
            #pragma unroll
            for (int r = 0; r < 8; ++r) {
                int n = node0 + mt * 16 + r + 8 * khalf;
                if (n < nN) {
                    float x = c[r] + (float)feat[n] * wsj + S[n] * vj;
                    emb_out[(size_t)n * EMB + j] = lrelu(x);
                }
            }
        }
    }
}

// -------- colsum[j] = sum_n emb[n][j] --------
__global__ void k_colsum(const float* __restrict__ emb, float* __restrict__ colsum, int nN) {
    int j = threadIdx.x;  // 128
    float acc = 0.0f;
    for (int n = blockIdx.x; n < nN; n += gridDim.x)
        acc += emb[(size_t)n * EMB + j];
    atomicAdd(&colsum[j], acc);
}

// -------- qconst = sum_j lrelu((w_q_allembed @ colsum)[j]) * w_q_reduc[j] --------
__global__ void k_qconst(const float* __restrict__ wqall, const float* __restrict__ colsum,
                         const float* __restrict__ wqreduc, float* __restrict__ qconst) {
    __shared__ float red[128];
    int j = threadIdx.x;  // 128
    float acc = 0.0f;
    #pragma unroll 4
    for (int k = 0; k < EMB; ++k) acc += wqall[j * EMB + k] * colsum[k];
    red[j] = lrelu(acc) * wqreduc[j];
    __syncthreads();
    for (int s = 64; s > 0; s >>= 1) {
        if (j < s) red[j] += red[j + s];
        __syncthreads();
    }
    if (j == 0) *qconst = red[0];
}

// -------- q[n] = qconst + sum_j lrelu((emb @ w_q_action^T)[n][j]) * r2[j] --------
__global__ __launch_bounds__(128) void k_qvals(
    const float* __restrict__ emb, const float* __restrict__ W,
    const float* __restrict__ r2, const float* __restrict__ qconst,
    float* __restrict__ qout, int nN)
{
    extern __shared__ float smem[];
    float* Wl   = smem;                             // [128][W_STRIDE] (TDM)
    float* At   = smem + EMB * W_STRIDE;            // [32][A_STRIDE]  (TDM, 1D padded)
    float* qacc = At + NODES_PER_BLOCK * A_STRIDE;  // [32]

    const int tid  = threadIdx.x;
    const int lane = tid & 31;
    const int wave = tid >> 5;
    const int node0 = blockIdx.x * NODES_PER_BLOCK;

    if (wave == 0) {
        tdm_load_padded(W, 0u, EMB, EMB, EMB, EMB, EMB);
        // 1D tile: 32 emb rows; tensor_dim0 bound makes OOB rows read as zero.
        unsigned remain = (unsigned)(nN - node0) * EMB;
        tdm_load_padded(emb + (size_t)node0 * EMB, (unsigned)(EMB * W_STRIDE * 4),
                        remain, 1u, NODES_PER_BLOCK * EMB, 0u, NODES_PER_BLOCK * EMB);
        __builtin_amdgcn_s_wait_tensorcnt(0);
    }
    if (tid < NODES_PER_BLOCK) qacc[tid] = 0.0f;
    __syncthreads();

    const int mrow  = lane & 15;
    const int khalf = lane >> 4;
    for (int tt = wave; tt < 16; tt += 4) {
        int mt = tt & 1;
        int jt = tt >> 1;
        v8f c = {};
        const float* ap = At + (mt * 16 + mrow) * A_STRIDE + 2 * khalf;
        const float* bp = Wl + (jt * 16 + mrow) * W_STRIDE + 2 * khalf;
        #pragma unroll 8
        for (int k0 = 0; k0 < EMB; k0 += 4) {
            v2f a = *(const v2f*)(ap + k0);
            v2f b = *(const v2f*)(bp + k0);
            c = __builtin_amdgcn_wmma_f32_16x16x4_f32(false, a, false, b,
                                                      (short)0, c, false, false);
        }
        int j = jt * 16 + mrow;
        float rj = r2[j];
        #pragma unroll
        for (int r = 0; r < 8; ++r) {
            int m = mt * 16 + r + 8 * khalf;
            if (node0 + m < nN) atomicAdd(&qacc[m], lrelu(c[r]) * rj);
        }
    }
    __syncthreads();
    if (tid < NODES_PER_BLOCK) {
        int n = node0 + tid;
        if (n < nN) qout[n] = qacc[tid] + *qconst;
    }
}

static inline size_t align16(size_t x) { return (x + 15) & ~size_t(15); }

extern "C" void kernel_launch(void* const* d_in, const int* in_sizes, int n_in,
                              void* d_out, int out_size, void* d_ws, size_t ws_size,
                              hipStream_t stream) {
    const int*   feat    = (const int*)d_in[0];
    const int*   esrc    = (const int*)d_in[1];
    const int*   edst    = (const int*)d_in[2];
    const float* ew      = (const float*)d_in[3];
    const float* wsel    = (const float*)d_in[4];
    const float* wnbp    = (const float*)d_in[5];
    const float* wnbw    = (const float*)d_in[6];
    const float* wew     = (const float*)d_in[7];
    const float* wqreduc = (const float*)d_in[8];
    const float* wqall   = (const float*)d_in[9];
    const float* wqact   = (const float*)d_in[10];
    const int nN = in_sizes[0];
    const int nE = in_sizes[1];

    // workspace layout
    char* p = (char*)d_ws;
    int*   rp     = (int*)p;    p += align16((size_t)(nN + 1) * sizeof(int));
    float* S      = (float*)p;  p += align16((size_t)nN * sizeof(float));
    float* v      = (float*)p;  p += align16(EMB * sizeof(float));
    float* colsum = (float*)p;  p += align16(EMB * sizeof(float));
    float* qconst = (float*)p;  p += align16(sizeof(float));
    float* emb_a  = (float*)p;  // nN*EMB floats

    float* qout      = (float*)d_out;
    float* emb_final = (float*)d_out + nN;   // final emb lives in the output buffer

    k_rowptr<<<(nN + 1 + 255) / 256, 256, 0, stream>>>(esrc, nE, nN, rp);
    k_nodewsum<<<(nN + 255) / 256, 256, 0, stream>>>(ew, rp, nN, S);
    k_precompute<<<1, 128, 0, stream>>>(wnbw, wew, v, colsum, qconst);
    k_emb1<<<(nN * EMB + 255) / 256, 256, 0, stream>>>(feat, S, wsel, v, emb_a, nN);

    const int gIter = (nN + NODES_PER_BLOCK - 1) / NODES_PER_BLOCK;
    const size_t ldsIter = (size_t)(EMB * W_STRIDE + NODES_PER_BLOCK * A_STRIDE) * sizeof(float);
    // iterations 2..4 (iter 1 handled by k_emb1 since emb starts at zero)
    k_iter<<<gIter, 128, ldsIter, stream>>>(emb_a, emb_final, rp, edst, feat, S, wsel, v, wnbp, nN);
    k_iter<<<gIter, 128, ldsIter, stream>>>(emb_final, emb_a, rp, edst, feat, S, wsel, v, wnbp, nN);
    k_iter<<<gIter, 128, ldsIter, stream>>>(emb_a, emb_final, rp, edst, feat, S, wsel, v, wnbp, nN);

    k_colsum<<<256, 128, 0, stream>>>(emb_final, colsum, nN);
    k_qconst<<<1, 128, 0, stream>>>(wqall, colsum, wqreduc, qconst);

    const size_t ldsQ = ldsIter + NODES_PER_BLOCK * sizeof(float);
    k_qvals<<<gIter, 128, ldsQ, stream>>>(emb_final, wqact, wqreduc + EMB, qconst, qout, nN);
}